// LocalGeometryAggregation_49108656063044
// MI455X (gfx1250) — compile-verified
//
#include <hip/hip_runtime.h>
#include <math.h>
#include <stdint.h>

typedef __attribute__((ext_vector_type(2))) float v2f;
typedef __attribute__((ext_vector_type(8))) float v8f;

#define B_   16
#define G_   512
#define K_   32
#define C_   64
#define OUTD 128
#define NPB  (G_ * K_)          // 16384 points per batch
#define NTOT (B_ * NPB)         // 262144 points total
#define EPSN 1e-5f

#if __has_builtin(__builtin_amdgcn_global_load_async_to_lds_b128)
#define HAVE_ASYNC_LDS 1
typedef int v4i __attribute__((vector_size(16)));
typedef __attribute__((address_space(1))) v4i as1_v4i;   // global
typedef __attribute__((address_space(3))) v4i as3_v4i;   // LDS
#else
#define HAVE_ASYNC_LDS 0
#endif

__device__ __forceinline__ void wait_async0() {
#if HAVE_ASYNC_LDS
#if __has_builtin(__builtin_amdgcn_s_wait_asynccnt)
    __builtin_amdgcn_s_wait_asynccnt(0);
#else
    asm volatile("s_wait_asynccnt 0x0" ::: "memory");
#endif
#endif
}

__device__ __forceinline__ float gelu_exact(float v) {
    return 0.5f * v * (1.0f + erff(v * 0.70710678118654752f));
}

// ---------------------------------------------------------------------------
// K1: x[b,c,g,k] = concat(knn_feats, center_feats) + sin/cos positional enc.
// One wave32 per (b,g): lanes = k. Wave-level max-reduce for coord scaling.
// ---------------------------------------------------------------------------
__global__ void prep_kernel(const float* __restrict__ lc,   // [B,G,3]
                            const float* __restrict__ lf,   // [B,G,64]
                            const float* __restrict__ kc,   // [B,G,32,3]
                            const float* __restrict__ kf,   // [B,G,32,64]
                            float* __restrict__ x)          // [B,128,G,K]
{
    const int lane = threadIdx.x & 31;
    const int wave = threadIdx.x >> 5;
    const int bg   = blockIdx.x * 8 + wave;           // [0, B*G)
    const int b    = bg / G_;
    const int g    = bg - b * G_;

    float rel[3];
#pragma unroll
    for (int d = 0; d < 3; ++d) {
        float r = kc[((size_t)bg * K_ + lane) * 3 + d] - lc[(size_t)bg * 3 + d];
        float a = fabsf(r);
#pragma unroll
        for (int m = 16; m >= 1; m >>= 1) a = fmaxf(a, __shfl_xor(a, m, 32));
        rel[d] = r / fmaxf(a, 1e-5f);
    }

    const size_t outbase = (size_t)b * OUTD * NPB + (size_t)g * K_ + lane;
    for (int c = 0; c < OUTD; ++c) {
        float feat = (c < C_) ? kf[((size_t)bg * K_ + lane) * C_ + c]
                              : lf[(size_t)bg * C_ + (c - C_)];
        const int d = c / 44;
        const int j = c - d * 44;
        const float jj  = (float)((j < 22) ? j : (j - 22));
        const float div = 100.0f * rel[d] / __powf(1000.0f, jj * (1.0f / 22.0f));
        const float pe  = (j < 22) ? __sinf(div) : __cosf(div);
        x[outbase + (size_t)c * NPB] = feat + pe;
    }
}

// ---------------------------------------------------------------------------
// K2: out[b,o,n] = sum_c W[o,c] * in[b,c,n]   (per-batch layout [Cin][16384])
// plus per-(b, group) sum / sumsq accumulation for GroupNorm.
//
// 256 threads = 8 waves. Each block processes 4 chunks of 64 points with
// DOUBLE-BUFFERED async global->LDS staging (GLOBAL_LOAD_ASYNC_TO_LDS_B128,
// ASYNCcnt-gated).  Weights are staged to LDS once, and each wave preloads
// the A fragments of its fixed 16-row o-tile into registers so the WMMA
// K-loop issues only B-fragment DS loads (one ds_load_2addr per WMMA).
// ---------------------------------------------------------------------------
template <int CIN, int COUT, int CPG>
__global__ void gemm_stats_kernel(const float* __restrict__ W,   // [COUT,CIN]
                                  const float* __restrict__ in,
                                  float* __restrict__ out,
                                  float* __restrict__ statSum,   // [B*8]
                                  float* __restrict__ statSq)    // [B*8]
{
    constexpr int O_TILES = COUT / 16;    // 4 or 8
    constexpr int WPO     = 8 / O_TILES;  // waves per o-tile: 2 or 1
    constexpr int NT      = 4 / WPO;      // n-tiles per wave: 2 or 4
    constexpr int NCHUNK  = 4;            // 64-pt chunks per block

    __shared__ __align__(16) float sW[COUT * (CIN + 1)];   // padded rows
    __shared__ __align__(16) float sIn[2][CIN * 64];

    const int t    = threadIdx.x;
    const int lane = t & 31;
    const int wave = t >> 5;
    const int row  = lane & 15;
    const int hi   = lane >> 4;                 // 0 or 1 (half-wave)

    const int n0g    = blockIdx.x * (64 * NCHUNK);  // global point index
    const int b      = n0g / NPB;
    const int n0base = n0g - b * NPB;               // within-batch base
    const float* inb  = in  + (size_t)b * CIN  * NPB;
    float*       outb = out + (size_t)b * COUT * NPB;

    // ---- async staging of one [CIN][64] tile ----------------------------
    auto issue_tile = [&](float* dst, int n0b) {
        for (int i = t; i < CIN * 16; i += 256) {
            const int c = i >> 4, q = i & 15;
            const float* src = &inb[(size_t)c * NPB + n0b + q * 4];
#if HAVE_ASYNC_LDS
            __builtin_amdgcn_global_load_async_to_lds_b128(
                (as1_v4i*)src, (as3_v4i*)&dst[c * 64 + q * 4], 0, 0);
#else
            *(float4*)&dst[c * 64 + q * 4] = *(const float4*)src;
#endif
        }
    };
    issue_tile(sIn[0], n0base);

    // ---- stage weights into LDS (row-padded against bank conflicts) -----
    for (int i = t; i < COUT * CIN; i += 256) {
        int o = i / CIN, c = i - o * CIN;
        sW[o * (CIN + 1) + c] = W[i];
    }
    // speculative prefetch of the next block's first chunk
    if (blockIdx.x + 1 < gridDim.x)
        __builtin_prefetch(&inb[(size_t)(t % CIN) * NPB + n0base + 64 * NCHUNK], 0, 1);

    wait_async0();
    __syncthreads();

    // ---- per-wave fixed o-tile; preload A fragments into registers ------
    const int otile = wave / WPO;
    const int sub   = wave % WPO;
    const int o0    = otile * 16;

    v2f areg[CIN / 4];
#pragma unroll
    for (int s = 0; s < CIN / 4; ++s) {
        areg[s].x = sW[(o0 + row) * (CIN + 1) + 4 * s + 2 * hi + 0];
        areg[s].y = sW[(o0 + row) * (CIN + 1) + 4 * s + 2 * hi + 1];
    }

    // ---- chunk loop with double-buffered async staging ------------------
    for (int it = 0; it < NCHUNK; ++it) {
        const int cur = it & 1;
        const int n0b = n0base + it * 64;
        if (it + 1 < NCHUNK)
            issue_tile(sIn[cur ^ 1], n0base + (it + 1) * 64);

        v8f acc[NT];
#pragma unroll
        for (int j = 0; j < NT; ++j)
            acc[j] = (v8f){0.f, 0.f, 0.f, 0.f, 0.f, 0.f, 0.f, 0.f};

#pragma unroll
        for (int s = 0; s < CIN / 4; ++s) {
            const int ci = 4 * s;
#pragma unroll
            for (int j = 0; j < NT; ++j) {
                const int n0l = (sub * NT + j) * 16;
                v2f bf;
                bf.x = sIn[cur][(ci + 2 * hi + 0) * 64 + n0l + row];
                bf.y = sIn[cur][(ci + 2 * hi + 1) * 64 + n0l + row];
                acc[j] = __builtin_amdgcn_wmma_f32_16x16x4_f32(
                    false, areg[s], false, bf, (short)0, acc[j], false, false);
            }
        }

        // store D tiles + group-norm statistics (each half-wave's 8 rows
        // fall in one channel group: CPG is 8 or 16, o0+hi*8 is 8-aligned)
        float s1 = 0.f, s2 = 0.f;
#pragma unroll
        for (int j = 0; j < NT; ++j) {
            const int n = n0b + (sub * NT + j) * 16 + row;
#pragma unroll
            for (int r = 0; r < 8; ++r) {
                const int o = o0 + hi * 8 + r;
                const float v = acc[j][r];
                outb[(size_t)o * NPB + n] = v;
                s1 += v; s2 += v * v;
            }
        }
#pragma unroll
        for (int m = 1; m < 16; m <<= 1) {      // reduce within each half-wave
            s1 += __shfl_xor(s1, m, 32);
            s2 += __shfl_xor(s2, m, 32);
        }
        if (row == 0) {
            const int grp = (o0 + hi * 8) / CPG;
            atomicAdd(&statSum[b * 8 + grp], s1);
            atomicAdd(&statSq [b * 8 + grp], s2);
        }

        if (it + 1 < NCHUNK) {
            wait_async0();      // next buffer's async loads have landed
            __syncthreads();    // everyone done reading the buffer we refill
        }
    }
}

// ---------------------------------------------------------------------------
// K3: finalize mean / rstd per (b, group)
// ---------------------------------------------------------------------------
__global__ void finalize_kernel(const float* __restrict__ sum,
                                const float* __restrict__ sq,
                                float* __restrict__ mean,
                                float* __restrict__ rstd,
                                float invCount)
{
    const int i = threadIdx.x;
    if (i < B_ * 8) {
        const float m = sum[i] * invCount;
        const float v = sq[i] * invCount - m * m;
        mean[i] = m;
        rstd[i] = rsqrtf(v + EPSN);
    }
}

// ---------------------------------------------------------------------------
// K4: h = gelu(groupnorm(h)) in place
// ---------------------------------------------------------------------------
template <int COUT, int CPG>
__global__ void norm_gelu_kernel(float* __restrict__ h,
                                 const float* __restrict__ mean,
                                 const float* __restrict__ rstd,
                                 const float* __restrict__ gamma,
                                 const float* __restrict__ beta)
{
    const size_t idx = (size_t)blockIdx.x * 256 + threadIdx.x;
    const size_t per = (size_t)COUT * NPB;
    const int b = (int)(idx / per);
    const size_t rem = idx - (size_t)b * per;
    const int o = (int)(rem / NPB);
    const int sg = b * 8 + o / CPG;
    const float v = (h[idx] - mean[sg]) * rstd[sg] * gamma[o] + beta[o];
    h[idx] = gelu_exact(v);
}

// ---------------------------------------------------------------------------
// K5: x = gelu(groupnorm(h2) + x) in place (residual close of a block)
// ---------------------------------------------------------------------------
__global__ void norm_resid_gelu_kernel(const float* __restrict__ h2,
                                       float* __restrict__ x,
                                       const float* __restrict__ mean,
                                       const float* __restrict__ rstd,
                                       const float* __restrict__ gamma,
                                       const float* __restrict__ beta)
{
    const size_t idx = (size_t)blockIdx.x * 256 + threadIdx.x;
    const size_t per = (size_t)OUTD * NPB;
    const int b = (int)(idx / per);
    const size_t rem = idx - (size_t)b * per;
    const int o = (int)(rem / NPB);
    const int sg = b * 8 + o / 16;              // 128 ch / 8 groups = 16 per grp
    const float v = (h2[idx] - mean[sg]) * rstd[sg] * gamma[o] + beta[o];
    x[idx] = gelu_exact(v + x[idx]);
}

// ---------------------------------------------------------------------------
extern "C" void kernel_launch(void* const* d_in, const int* in_sizes, int n_in,
                              void* d_out, int out_size, void* d_ws, size_t ws_size,
                              hipStream_t stream)
{
    const float* lc = (const float*)d_in[0];
    const float* lf = (const float*)d_in[1];
    const float* kc = (const float*)d_in[2];
    const float* kf = (const float*)d_in[3];

    float* x  = (float*)d_out;                            // [B,128,G,K]
    float* ws = (float*)d_ws;
    float* h    = ws;                                     // 64  * NTOT floats
    float* h2   = ws + (size_t)64  * NTOT;                // 128 * NTOT floats
    float* stat = ws + (size_t)192 * NTOT;                // 512 floats
    float* sum  = stat;
    float* sq   = stat + 128;
    float* mean = stat + 256;
    float* rstd = stat + 384;

    prep_kernel<<<(B_ * G_) / 8, 256, 0, stream>>>(lc, lf, kc, kf, x);

    for (int blk = 0; blk < 3; ++blk) {
        const float* W1 = (const float*)d_in[4 + blk * 6 + 0];  // [64,128]
        const float* g1 = (const float*)d_in[4 + blk * 6 + 1];
        const float* b1 = (const float*)d_in[4 + blk * 6 + 2];
        const float* W2 = (const float*)d_in[4 + blk * 6 + 3];  // [128,64]
        const float* g2 = (const float*)d_in[4 + blk * 6 + 4];
        const float* b2 = (const float*)d_in[4 + blk * 6 + 5];

        (void)hipMemsetAsync(stat, 0, 256 * sizeof(float), stream);
        gemm_stats_kernel<128, 64, 8><<<NTOT / 256, 256, 0, stream>>>(W1, x, h, sum, sq);
        finalize_kernel<<<1, 128, 0, stream>>>(sum, sq, mean, rstd,
                                               1.0f / (8.0f * NPB));
        norm_gelu_kernel<64, 8><<<(64u * NTOT) / 256, 256, 0, stream>>>(
            h, mean, rstd, g1, b1);

        (void)hipMemsetAsync(stat, 0, 256 * sizeof(float), stream);
        gemm_stats_kernel<64, 128, 16><<<NTOT / 256, 256, 0, stream>>>(W2, h, h2, sum, sq);
        finalize_kernel<<<1, 128, 0, stream>>>(sum, sq, mean, rstd,
                                               1.0f / (16.0f * NPB));
        norm_resid_gelu_kernel<<<(128u * NTOT) / 256, 256, 0, stream>>>(
            h2, x, mean, rstd, g2, b2);
    }
}